// SimSiamLoss_8486855377129
// MI455X (gfx1250) — compile-verified
//
#include <hip/hip_runtime.h>
#include <hip/hip_bf16.h>

typedef __attribute__((ext_vector_type(2))) float v2f;
typedef __attribute__((ext_vector_type(8))) float v8f;

#define N_ROWS 8192
#define DIM    128
#define NCLS   512
#define BLK1   256
#define ROWS_PER_BLK (BLK1 / 32)           // 8 rows (waves) per block
#define NBLK1  (N_ROWS / ROWS_PER_BLK)     // 1024 blocks
#define WS_WORDS (2 * NCLS * DIM + NCLS + NBLK1)

// ---------------------------------------------------------------- zero ws
__global__ void simsiam_zero_ws(float* __restrict__ ws, int n) {
    int i = blockIdx.x * blockDim.x + threadIdx.x;
    if (i < n) ws[i] = 0.0f;
}

// --------------------------------------------- normalize + scatter-add
// One wave32 per row: 4 floats/lane. Reduce ||p||^2, ||z||^2, p.z across
// the wave, then atomically accumulate normalized rows into per-class sums.
__global__ __launch_bounds__(BLK1)
void simsiam_norm_scatter(const float* __restrict__ ps,
                          const float* __restrict__ zs,
                          const int*   __restrict__ tgt,
                          float* __restrict__ P,
                          float* __restrict__ Z,
                          unsigned* __restrict__ cnt,
                          float* __restrict__ diagBuf) {
    __shared__ float sdiag;
    const int lane = threadIdx.x & 31;
    const int wid  = threadIdx.x >> 5;
    const int row  = blockIdx.x * ROWS_PER_BLK + wid;   // grid exactly covers rows

    if (threadIdx.x == 0) sdiag = 0.0f;
    __syncthreads();

    const float4 pv = ((const float4*)(ps + (size_t)row * DIM))[lane];
    const float4 zv = ((const float4*)(zs + (size_t)row * DIM))[lane];

    float pp = pv.x*pv.x + pv.y*pv.y + pv.z*pv.z + pv.w*pv.w;
    float zz = zv.x*zv.x + zv.y*zv.y + zv.z*zv.z + zv.w*zv.w;
    float pz = pv.x*zv.x + pv.y*zv.y + pv.z*zv.z + pv.w*zv.w;

    #pragma unroll
    for (int m = 16; m >= 1; m >>= 1) {
        pp += __shfl_xor(pp, m, 32);
        zz += __shfl_xor(zz, m, 32);
        pz += __shfl_xor(pz, m, 32);
    }

    const float rp = 1.0f / fmaxf(sqrtf(pp), 1e-8f);
    const float rz = 1.0f / fmaxf(sqrtf(zz), 1e-8f);
    const int   c  = tgt[row];

    float* pd = P + (size_t)c * DIM + lane * 4;
    atomicAdd(pd + 0, pv.x * rp);
    atomicAdd(pd + 1, pv.y * rp);
    atomicAdd(pd + 2, pv.z * rp);
    atomicAdd(pd + 3, pv.w * rp);

    float* zd = Z + (size_t)c * DIM + lane * 4;
    atomicAdd(zd + 0, zv.x * rz);
    atomicAdd(zd + 1, zv.y * rz);
    atomicAdd(zd + 2, zv.z * rz);
    atomicAdd(zd + 3, zv.w * rz);

    if (lane == 0) {
        atomicAdd(&sdiag, pz * rp * rz);   // pn_i . zn_i
        atomicAdd(&cnt[c], 1u);
    }
    __syncthreads();
    if (threadIdx.x == 0) diagBuf[blockIdx.x] = sdiag;
}

// ------------------------------------------------------------- finalize
// One 1024-thread block (32 waves). Wave w computes trace(P_w @ Z_w^T) for
// its 16-class tile with V_WMMA_F32_16X16X4_F32 chained over K=128.
__global__ __launch_bounds__(1024)
void simsiam_finalize(const float* __restrict__ P,
                      const float* __restrict__ Z,
                      const unsigned* __restrict__ cnt,
                      const float* __restrict__ diagBuf,
                      float* __restrict__ out) {
    __shared__ float    sS;
    __shared__ float    sDiag;
    __shared__ unsigned sPairs;

    const int tid  = threadIdx.x;
    const int lane = tid & 31;
    const int wave = tid >> 5;   // 0..31 -> class tile

    if (tid == 0) { sS = 0.0f; sDiag = 0.0f; sPairs = 0u; }
    __syncthreads();

    // npairs = sum_c m_c*(m_c-1)/2
    if (tid < NCLS) {
        unsigned m = cnt[tid];
        atomicAdd(&sPairs, (m * (m - 1u)) / 2u);
    }

    // reduce per-block diag partials (NBLK1 == blockDim == 1024)
    {
        float v = diagBuf[tid];
        #pragma unroll
        for (int m = 16; m >= 1; m >>= 1) v += __shfl_xor(v, m, 32);
        if (lane == 0) atomicAdd(&sDiag, v);
    }

    // WMMA block trace: D = P_tile(16xK) @ Z_tile^T(Kx16), accumulate diag.
    // 32-bit operand layout: lane l, vgpr v <-> element [l&15][2*(l>>4)+v],
    // identical for A and B, so loading Z rows in A-pattern supplies Z^T as B.
    {
        const int   mrow = 16 * wave + (lane & 15);
        const int   koff = 2 * (lane >> 4);
        const float* prow = P + (size_t)mrow * DIM + koff;
        const float* zrow = Z + (size_t)mrow * DIM + koff;

        v8f acc = {0.f, 0.f, 0.f, 0.f, 0.f, 0.f, 0.f, 0.f};
        #pragma unroll
        for (int k = 0; k < DIM; k += 4) {
            v2f a = *(const v2f*)(prow + k);
            v2f b = *(const v2f*)(zrow + k);
            acc = __builtin_amdgcn_wmma_f32_16x16x4_f32(
                false, a, false, b, (short)0, acc, false, false);
        }
        // diag of 16x16 D: vgpr j holds D[j][lane] (lanes 0-15) and
        // D[8+j][lane-16] (lanes 16-31) -> D[j][j] at lane j, D[8+j][8+j] at 24+j
        float dsum = 0.0f;
        #pragma unroll
        for (int j = 0; j < 8; ++j)
            if (lane == j || lane == 24 + j) dsum += acc[j];
        #pragma unroll
        for (int m = 16; m >= 1; m >>= 1) dsum += __shfl_xor(dsum, m, 32);
        if (lane == 0) atomicAdd(&sS, dsum);
    }

    __syncthreads();
    if (tid == 0) {
        float npairs = fmaxf((float)sPairs, 1.0f);
        out[0] = -0.5f * (sS - sDiag) / npairs;
    }
}

// --------------------------------------------------------------- launch
extern "C" void kernel_launch(void* const* d_in, const int* in_sizes, int n_in,
                              void* d_out, int out_size, void* d_ws, size_t ws_size,
                              hipStream_t stream) {
    (void)in_sizes; (void)n_in; (void)out_size; (void)ws_size;

    const float* ps  = (const float*)d_in[0];
    const float* zs  = (const float*)d_in[1];
    const int*   tgt = (const int*)d_in[2];
    float*       out = (float*)d_out;

    float*    P       = (float*)d_ws;
    float*    Z       = P + (size_t)NCLS * DIM;
    unsigned* cnt     = (unsigned*)(Z + (size_t)NCLS * DIM);
    float*    diagBuf = (float*)(cnt + NCLS);

    const int nwords = WS_WORDS;
    simsiam_zero_ws<<<(nwords + 255) / 256, 256, 0, stream>>>((float*)d_ws, nwords);
    simsiam_norm_scatter<<<NBLK1, BLK1, 0, stream>>>(ps, zs, tgt, P, Z, cnt, diagBuf);
    simsiam_finalize<<<1, 1024, 0, stream>>>(P, Z, cnt, diagBuf, out);
}